// XDGRASP_24111946400512
// MI455X (gfx1250) — compile-verified
//
#include <hip/hip_runtime.h>

// ---------------- problem constants ----------------
#define T_   2
#define PH_  4
#define CH_  8
#define H_   128
#define W_   128
#define K_   1024
#define HW_  (H_ * W_)
#define TP_  (T_ * PH_)

#define LAM1_ 0.01f
#define LAM2_ 0.01f
#define LR_   0.5f
// mean denominators
#define INV_N2_ (1.0f / 131072.0f)   // T*PH*CH*D*K*2 real elements in loss_dc
#define INV_NC_ (1.0f / 65536.0f)    // (T-1)*PH*D*H*W complex elements in tv_c
#define INV_NR_ (1.0f / 98304.0f)    // T*(PH-1)*D*H*W complex elements in tv_r

typedef float v2f __attribute__((ext_vector_type(2)));
typedef float v8f __attribute__((ext_vector_type(8)));
typedef unsigned int v4u __attribute__((ext_vector_type(4)));
typedef int v4i __attribute__((ext_vector_type(4)));
typedef int v8i __attribute__((ext_vector_type(8)));

// D = A(16x4) * B(4x16) + C, fp32 WMMA (no A/B neg modifiers exist for F32;
// subtractions are folded by negating operand data in VALU).
static __device__ __forceinline__ v8f wmma4(v2f a, v2f b, v8f c) {
    return __builtin_amdgcn_wmma_f32_16x16x4_f32(
        /*neg_a=*/false, a, /*neg_b=*/false, b,
        /*c_mod=*/(short)0, c, /*reuse_a=*/false, /*reuse_b=*/false);
}

// ---------------- Ex / Ey exponential tables (planar re/im) ----------------
// Ex[tp][k][h] = exp(-i * traj_x[tp,k] * (h - H/2)); Ey analogous with traj_y.
__global__ __launch_bounds__(256) void k_exp(const float* __restrict__ traj,
                                             float* __restrict__ Exr, float* __restrict__ Exi,
                                             float* __restrict__ Eyr, float* __restrict__ Eyi) {
    int i = blockIdx.x * blockDim.x + threadIdx.x;   // 2 * TP * K * 128 threads
    int n1 = TP_ * K_ * H_;
    if (i >= 2 * n1) return;
    int axis = i / n1;            // 0 -> Ex, 1 -> Ey
    int r = i - axis * n1;
    int h = r % H_;
    int tpk = r / H_;
    int k = tpk % K_;
    int tp = tpk / K_;
    float coord = traj[(tp * 2 + axis) * K_ + k];
    float x = coord * (float)(h - H_ / 2);
    float s, c;
    __sincosf(x, &s, &c);
    if (axis == 0) { Exr[r] = c; Exi[r] = -s; }
    else           { Eyr[r] = c; Eyi[r] = -s; }
}

// ---------------- csm normalization (interleaved complex) ----------------
__global__ __launch_bounds__(256) void k_csm(const float* __restrict__ csm,
                                             float* __restrict__ out) {
    int i = blockIdx.x * blockDim.x + threadIdx.x;   // T*CH*HW
    if (i >= T_ * CH_ * HW_) return;
    float cr = csm[2 * i], ci = csm[2 * i + 1];
    float inv = rsqrtf(cr * cr + ci * ci + 1e-30f);
    out[2 * i] = cr * inv;
    out[2 * i + 1] = ci * inv;
}

// ---------------- img_mc = p * csm_n  (planar output) ----------------
__global__ __launch_bounds__(256) void k_mulcsm(const float* __restrict__ p,
                                                const float* __restrict__ csm,
                                                float* __restrict__ imr,
                                                float* __restrict__ imi) {
    int i = blockIdx.x * blockDim.x + threadIdx.x;   // T*PH*CH*HW
    if (i >= T_ * PH_ * CH_ * HW_) return;
    int hw = i % HW_;
    int c  = (i / HW_) % CH_;
    int ph = (i / (HW_ * CH_)) % PH_;
    int t  = i / (HW_ * CH_ * PH_);
    int pi_ = ((t * PH_ + ph) * HW_ + hw) * 2;
    int ci_ = ((t * CH_ + c) * HW_ + hw) * 2;
    float pr = p[pi_], pim = p[pi_ + 1];
    float cr = csm[ci_], cim = csm[ci_ + 1];
    imr[i] = pr * cr - pim * cim;
    imi[i] = pr * cim + pim * cr;
}

// ---------------- forward NUFFT + residual ----------------
// One wave handles one (t,ph,ch) instance and a 16-row k tile.
// C(k,w) = Sum_h Ex[k,h]*Img[h,w] via WMMA, then est[k] += Sum_w C(k,w)*Ey[k,w],
// diff = est - ksp.
__global__ __launch_bounds__(256) void k_forward(const float* __restrict__ imr,
                                                 const float* __restrict__ imi,
                                                 const float* __restrict__ Exr,
                                                 const float* __restrict__ Exi,
                                                 const float* __restrict__ Eyr,
                                                 const float* __restrict__ Eyi,
                                                 const float* __restrict__ ksp,
                                                 float* __restrict__ diff) {
    int lane = threadIdx.x & 31;
    int wid = (blockIdx.x * blockDim.x + threadIdx.x) >> 5;   // 4096 waves
    int kt = wid & 63;          // k tile
    int inst = wid >> 6;        // (t*PH+ph)*CH + ch
    int tp = inst >> 3;
    int l16 = lane & 15;
    int lh = lane >> 4;
    int k0 = kt * 16;

    const float* exr = Exr + tp * K_ * H_;
    const float* exi = Exi + tp * K_ * H_;
    const float* eyr = Eyr + tp * K_ * W_;
    const float* eyi = Eyi + tp * K_ * W_;
    const float* ir = imr + inst * HW_;
    const float* ii = imi + inst * HW_;

    // hint the image tile into cache ahead of the streaming loop
    __builtin_prefetch(ir + lane * 4, 0, 0);
    __builtin_prefetch(ii + lane * 4, 0, 0);

    float estr[8], esti[8];
#pragma unroll
    for (int j = 0; j < 8; ++j) { estr[j] = 0.f; esti[j] = 0.f; }

    int arow = (k0 + l16) * H_;   // A: M = k row per lane (lanes 0-15 / 16-31 same M)

    for (int wt = 0; wt < 8; ++wt) {
        int w0 = wt * 16 + l16;
        v8f Cr = {}; v8f Ci = {};
        for (int hc = 0; hc < 32; ++hc) {
            int hA = hc * 4 + 2 * lh;   // K-slot pair per half-wave
            v2f Ar, Ai, Br, Bi, nBi;
            Ar.x = exr[arow + hA];       Ar.y = exr[arow + hA + 1];
            Ai.x = exi[arow + hA];       Ai.y = exi[arow + hA + 1];
            Br.x = ir[hA * W_ + w0];     Br.y = ir[(hA + 1) * W_ + w0];
            Bi.x = ii[hA * W_ + w0];     Bi.y = ii[(hA + 1) * W_ + w0];
            nBi.x = -Bi.x;               nBi.y = -Bi.y;
            // C = Ex * Img (complex): Cr = ExR*Ir - ExI*Ii ; Ci = ExR*Ii + ExI*Ir
            Cr = wmma4(Ar, Br, Cr);
            Cr = wmma4(Ai, nBi, Cr);
            Ci = wmma4(Ar, Bi, Ci);
            Ci = wmma4(Ai, Br, Ci);
        }
        // est[k] += Sum_w C(k,w) * Ey(k,w): 16-lane butterfly per half-wave
#pragma unroll
        for (int j = 0; j < 8; ++j) {
            int krow = k0 + j + 8 * lh;
            float er = eyr[krow * W_ + w0];
            float ei = eyi[krow * W_ + w0];
            float cr = Cr[j], ci = Ci[j];
            float pr = cr * er - ci * ei;
            float pm = cr * ei + ci * er;
#pragma unroll
            for (int m = 1; m < 16; m <<= 1) {
                pr += __shfl_xor(pr, m, 32);
                pm += __shfl_xor(pm, m, 32);
            }
            estr[j] += pr;
            esti[j] += pm;
        }
    }

    if (l16 == 0) {   // lanes 0 and 16 own rows k0..k0+7 / k0+8..k0+15
        const float* y = ksp + (inst * K_ + k0 + 8 * lh) * 2;
        float* o = diff + (inst * K_ + k0 + 8 * lh) * 2;
#pragma unroll
        for (int j = 0; j < 8; ++j) {
            o[2 * j]     = estr[j] - y[2 * j];
            o[2 * j + 1] = esti[j] - y[2 * j + 1];
        }
    }
}

// ---------------- adjoint NUFFT + coil combine ----------------
// One wave handles one (t,ph) and a 16x16 (h,w) tile, all 8 coils (channels inner
// so the Ex^H A-tile is reused 8x per K chunk). All 8 waves of a block share the
// same (t,ph), so the 64KB residual block d[tp,:,:] is staged into LDS once per
// block by the Tensor Data Mover (1-D tile descriptor, 8192 x 8B elements), then
// consumed via ds_load_b128 in the K loop. Epilogue fuses conj(csm) coil combine.
__global__ __launch_bounds__(256) void k_adjoint(const float* __restrict__ d,
                                                 const float* __restrict__ Exr,
                                                 const float* __restrict__ Exi,
                                                 const float* __restrict__ Eyr,
                                                 const float* __restrict__ Eyi,
                                                 const float* __restrict__ csm,
                                                 float* __restrict__ out,
                                                 float scale) {
    int lane = threadIdx.x & 31;
    int wid = (blockIdx.x * blockDim.x + threadIdx.x) >> 5;   // 512 waves
    int wt = wid & 7;
    int ht = (wid >> 3) & 7;
    int tp = wid >> 6;
    int t = tp / PH_;
    int l16 = lane & 15;
    int lh = lane >> 4;
    int h0 = ht * 16, w0 = wt * 16;

    // -------- TDM: stage d[tp, 0:CH, 0:K] (8192 complex = 64KB) into LDS --------
    __shared__ __align__(16) float smem[CH_ * K_ * 2];
    {
        const float* dblk = d + (size_t)tp * (CH_ * K_ * 2);
        unsigned long long ga = (unsigned long long)(uintptr_t)dblk;
        unsigned int lds = (unsigned int)(uintptr_t)smem;   // addr[31:0] = LDS byte offset
        const unsigned int NEL = CH_ * K_;                  // 8192 elements of 8 bytes
        v4u g0;
        g0[0] = 1u;                                         // count=1, user descriptor
        g0[1] = lds;                                        // lds_addr (bytes)
        g0[2] = (unsigned int)(ga & 0xFFFFFFFFu);           // global_addr[31:0]
        g0[3] = (unsigned int)((ga >> 32) & 0x01FFFFFFu)    // global_addr[56:32]
                | 0x80000000u;                              // type=2 ("image")
        v8i g1;
        g1[0] = (int)(3u << 16);                            // data_size = 8B
        g1[1] = (int)((NEL & 0xFFFFu) << 16);               // tensor_dim0[15:0]
        g1[2] = (int)((NEL >> 16) | (1u << 16));            // tensor_dim0[31:16] | tensor_dim1=1
        g1[3] = (int)((NEL & 0xFFFFu) << 16);               // tile_dim0 = 8192
        g1[4] = (int)1u;                                    // tile_dim1 = 1 (tile_dim2 unused)
        g1[5] = (int)NEL;                                   // tensor_dim0_stride[31:0]
        g1[6] = 0;
        g1[7] = 0;
        v4i gz4 = {0, 0, 0, 0};                             // <=2D: groups 2/3 unused
        v8i gz8 = {0, 0, 0, 0, 0, 0, 0, 0};
        if ((threadIdx.x >> 5) == 0) {                      // wave 0 issues the DMA
            __builtin_amdgcn_tensor_load_to_lds(g0, g1, gz4, gz4, gz8, 0);
            __builtin_amdgcn_s_wait_tensorcnt(0);
        }
        __syncthreads();
    }
    const float4* dl = (const float4*)smem;   // [(c*K+k)/2] -> (re0,im0,re1,im1)

    const float* exr = Exr + tp * K_ * H_;
    const float* exi = Exi + tp * K_ * H_;
    const float* eyr = Eyr + tp * K_ * W_;
    const float* eyi = Eyi + tp * K_ * W_;

    v8f accR[CH_], accI[CH_];
#pragma unroll
    for (int c = 0; c < CH_; ++c) { accR[c] = (v8f){}; accI[c] = (v8f){}; }

    for (int kc = 0; kc < K_ / 4; ++kc) {
        int ks = kc * 4 + 2 * lh;
        v2f Ar, Ai;
        Ar.x = exr[ks * H_ + h0 + l16];       Ar.y = exr[(ks + 1) * H_ + h0 + l16];
        Ai.x = exi[ks * H_ + h0 + l16];       Ai.y = exi[(ks + 1) * H_ + h0 + l16];
        float er0 = eyr[ks * W_ + w0 + l16];  float er1 = eyr[(ks + 1) * W_ + w0 + l16];
        float ei0 = eyi[ks * W_ + w0 + l16];  float ei1 = eyi[(ks + 1) * W_ + w0 + l16];
#pragma unroll
        for (int c = 0; c < CH_; ++c) {
            const float4 dd = dl[(c * K_ + ks) >> 1];   // two consecutive complex samples
            // M = d * conj(Ey)
            float Mr0 = dd.x * er0 + dd.y * ei0;
            float Mi0 = dd.y * er0 - dd.x * ei0;
            float Mr1 = dd.z * er1 + dd.w * ei1;
            float Mi1 = dd.w * er1 - dd.z * ei1;
            v2f Br, Bi, nBr;
            Br.x = Mr0;  Br.y = Mr1;
            Bi.x = Mi0;  Bi.y = Mi1;
            nBr.x = -Mr0; nBr.y = -Mr1;
            // G = conj(Ex)^T * M: Gr = ExR*Mr + ExI*Mi ; Gi = ExR*Mi - ExI*Mr
            accR[c] = wmma4(Ar, Br, accR[c]);
            accR[c] = wmma4(Ai, Bi, accR[c]);
            accI[c] = wmma4(Ar, Bi, accI[c]);
            accI[c] = wmma4(Ai, nBr, accI[c]);
        }
    }

    float sr[8], si[8];
#pragma unroll
    for (int j = 0; j < 8; ++j) { sr[j] = 0.f; si[j] = 0.f; }
#pragma unroll
    for (int c = 0; c < CH_; ++c) {
        int base = (t * CH_ + c) * HW_;
#pragma unroll
        for (int j = 0; j < 8; ++j) {
            int h = h0 + j + 8 * lh;
            int w = w0 + l16;
            const float2 cs = *(const float2*)(csm + (base + h * W_ + w) * 2);
            float gr = accR[c][j], gi = accI[c][j];
            sr[j] += cs.x * gr + cs.y * gi;   // conj(csm) * G
            si[j] += cs.x * gi - cs.y * gr;
        }
    }
#pragma unroll
    for (int j = 0; j < 8; ++j) {
        int h = h0 + j + 8 * lh;
        int idx = (tp * HW_ + h * W_ + w0 + l16) * 2;
        out[idx] = scale * sr[j];
        out[idx + 1] = scale * si[j];
    }
}

// ---------------- TV gradient + gradient step (double buffered) ----------------
__global__ __launch_bounds__(256) void k_update(const float* __restrict__ pin,
                                                const float* __restrict__ g,
                                                float* __restrict__ pout) {
    int i = blockIdx.x * blockDim.x + threadIdx.x;   // T*PH*HW
    if (i >= T_ * PH_ * HW_) return;
    int ph = (i / HW_) % PH_;
    int t = i / (HW_ * PH_);
    float pr = pin[2 * i], pm = pin[2 * i + 1];
    float tr = 0.f, ti = 0.f;

    auto unit = [](float dr, float di, float& ur, float& ui) {
        float m = sqrtf(dr * dr + di * di);
        float inv = (m > 1e-20f) ? 1.f / m : 0.f;
        ur = dr * inv; ui = di * inv;
    };
    float ur, ui;
    if (t + 1 < T_) {  // diff starting here: p[t+1]-p[t]
        int j = i + PH_ * HW_;
        unit(pin[2 * j] - pr, pin[2 * j + 1] - pm, ur, ui);
        tr -= LAM1_ * INV_NC_ * ur;  ti -= LAM1_ * INV_NC_ * ui;
    }
    if (t > 0) {       // diff ending here: p[t]-p[t-1]
        int j = i - PH_ * HW_;
        unit(pr - pin[2 * j], pm - pin[2 * j + 1], ur, ui);
        tr += LAM1_ * INV_NC_ * ur;  ti += LAM1_ * INV_NC_ * ui;
    }
    if (ph + 1 < PH_) {
        int j = i + HW_;
        unit(pin[2 * j] - pr, pin[2 * j + 1] - pm, ur, ui);
        tr -= LAM2_ * INV_NR_ * ur;  ti -= LAM2_ * INV_NR_ * ui;
    }
    if (ph > 0) {
        int j = i - HW_;
        unit(pr - pin[2 * j], pm - pin[2 * j + 1], ur, ui);
        tr += LAM2_ * INV_NR_ * ur;  ti += LAM2_ * INV_NR_ * ui;
    }
    pout[2 * i]     = pr - LR_ * (g[2 * i] + tr);
    pout[2 * i + 1] = pm - LR_ * (g[2 * i + 1] + ti);
}

// ---------------- host orchestration ----------------
extern "C" void kernel_launch(void* const* d_in, const int* in_sizes, int n_in,
                              void* d_out, int out_size, void* d_ws, size_t ws_size,
                              hipStream_t stream) {
    (void)in_sizes; (void)n_in; (void)out_size; (void)ws_size;
    const float* ksp  = (const float*)d_in[0];   // kspace_data
    const float* kspc = (const float*)d_in[1];   // kspace_data_compensated
    const float* traj = (const float*)d_in[2];   // kspace_traj
    const float* csm  = (const float*)d_in[3];   // csm
    float* out = (float*)d_out;
    float* ws = (float*)d_ws;

    // workspace layout (float offsets)
    float* EXR  = ws;                              // 1048576
    float* EXI  = EXR + 1048576;
    float* EYR  = EXI + 1048576;
    float* EYI  = EYR + 1048576;
    float* CSM  = EYI + 1048576;                   // 524288 (interleaved)
    float* P    = CSM + 524288;                    // 262144 (interleaved)
    float* P2   = P + 262144;                      // 262144
    float* G    = P2 + 262144;                     // 262144
    float* IMR  = G + 262144;                      // 1048576 (planar)
    float* IMI  = IMR + 1048576;                   // 1048576
    float* DIFF = IMI + 1048576;                   // 131072 (interleaved)

    const int NPIX = T_ * PH_ * HW_ * 2;           // 262144 floats (complex interleaved)
    float* img_init_out = out + NPIX;              // second output region

    k_exp<<<(2 * TP_ * K_ * H_) / 256, 256, 0, stream>>>(traj, EXR, EXI, EYR, EYI);
    k_csm<<<(T_ * CH_ * HW_) / 256, 256, 0, stream>>>(csm, CSM);

    // image_init = adjoint(compensated) coil-combined / 5 -> straight into d_out
    k_adjoint<<<64, 256, 0, stream>>>(kspc, EXR, EXI, EYR, EYI, CSM, img_init_out, 0.2f);
    (void)hipMemcpyAsync(P, img_init_out, (size_t)NPIX * sizeof(float),
                         hipMemcpyDeviceToDevice, stream);

    float* pcur = P;
    float* pnext = P2;
    for (int step = 0; step < 2; ++step) {
        k_mulcsm<<<(T_ * PH_ * CH_ * HW_) / 256, 256, 0, stream>>>(pcur, CSM, IMR, IMI);
        k_forward<<<512, 256, 0, stream>>>(IMR, IMI, EXR, EXI, EYR, EYI, ksp, DIFF);
        k_adjoint<<<64, 256, 0, stream>>>(DIFF, EXR, EXI, EYR, EYI, CSM, G, INV_N2_);
        k_update<<<(T_ * PH_ * HW_) / 256, 256, 0, stream>>>(pcur, G, pnext);
        float* tmp = pcur; pcur = pnext; pnext = tmp;
    }
    (void)hipMemcpyAsync(out, pcur, (size_t)NPIX * sizeof(float),
                         hipMemcpyDeviceToDevice, stream);
}